// VTSNE_70093866270910
// MI455X (gfx1250) — compile-verified
//
#include <hip/hip_runtime.h>
#include <hip/hip_bf16.h>
#include <stdint.h>

// ---------------------------------------------------------------------------
// VTSNE loss on MI455X (gfx1250, wave32).
// Partition function: 8192x8192 Student-t kernel sum via Gram-matrix identity
//   d2(i,j) = |xi|^2 + |xj|^2 - 2 xi.xj   (T=2 topics, K zero-padded to 4)
// One V_WMMA_F32_16X16X4_F32 per 16x16 tile; column panel staged to LDS via
// gfx1250 async-to-LDS loads (ASYNCcnt) and read back with ds_load.
// ---------------------------------------------------------------------------

typedef __attribute__((ext_vector_type(2))) float v2f;
typedef __attribute__((ext_vector_type(4))) float v4f;
typedef __attribute__((ext_vector_type(8))) float v8f;

#define N_POINTS   8192
#define N_PAIRS    65536
#define TILE_BLKS  4096     // (8192/128)^2
#define PTS_BLKS   32       // 8192/256
#define PAIR_BLKS  256      // 65536/256

// workspace layout (float offsets)
#define WS_XY   0                              // 8192 float2 = 16384 floats
#define WS_NRM  (WS_XY + 2 * N_POINTS)         // 8192 floats
#define WS_KLD  (WS_NRM + N_POINTS)            // 32 partials
#define WS_TILE (WS_KLD + PTS_BLKS)            // 4096 partials
#define WS_S1   (WS_TILE + TILE_BLKS)          // 256 partials
#define WS_SP   (WS_S1 + PAIR_BLKS)            // 256 partials

__device__ __forceinline__ uint32_t pcg_hash(uint32_t v) {
    v = v * 747796405u + 2891336453u;
    uint32_t w = ((v >> ((v >> 28u) + 4u)) ^ v) * 277803737u;
    return (w >> 22u) ^ w;
}

__device__ __forceinline__ float hash_gauss(uint32_t idx) {
    uint32_t h1 = pcg_hash(idx * 2u + 0x2A6A2A2Au);
    uint32_t h2 = pcg_hash(idx * 2u + 1u + 0x2A6A2A2Au);
    float u1 = ((float)(h1 >> 8) + 0.5f) * (1.0f / 16777216.0f);  // (0,1)
    float u2 = ((float)(h2 >> 8) + 0.5f) * (1.0f / 16777216.0f);
    return sqrtf(-2.0f * logf(u1)) * cosf(6.28318530718f * u2);
}

__device__ __forceinline__ float wave_reduce(float v) {
    #pragma unroll
    for (int o = 16; o > 0; o >>= 1) v += __shfl_xor(v, o, 32);
    return v;
}

// --------------------------- K1: points ------------------------------------
__global__ void k_points(const float* __restrict__ mu, const float* __restrict__ lv,
                         float2* __restrict__ xy, float* __restrict__ nrm,
                         float* __restrict__ kldPart) {
    __shared__ float sh[8];
    int p = blockIdx.x * 256 + threadIdx.x;
    float m0 = mu[2 * p],  m1 = mu[2 * p + 1];
    float l0 = lv[2 * p],  l1 = lv[2 * p + 1];
    float e0 = expf(l0), e1 = expf(l1);
    float s0 = expf(0.5f * l0), s1 = expf(0.5f * l1);
    float xa = m0 + hash_gauss((uint32_t)(2 * p))     * s0;
    float xb = m1 + hash_gauss((uint32_t)(2 * p + 1)) * s1;
    xy[p]  = make_float2(xa, xb);
    nrm[p] = xa * xa + xb * xb;
    float kld = (1.0f + l0 - m0 * m0 - e0) + (1.0f + l1 - m1 * m1 - e1);

    kld = wave_reduce(kld);
    int lane = threadIdx.x & 31, wid = threadIdx.x >> 5;
    if (lane == 0) sh[wid] = kld;
    __syncthreads();
    if (threadIdx.x == 0) {
        float s = 0.f;
        #pragma unroll
        for (int k = 0; k < 8; ++k) s += sh[k];
        kldPart[blockIdx.x] = s;
    }
}

// --------------------------- K2: WMMA tiles ---------------------------------
// grid (64,64), 256 threads (8 waves). Block = 128x128 macro-tile; each wave
// owns a 16-row stripe. Column panel (128 float2 + 128 norms) staged in LDS
// once per block via async-to-LDS, shared by all 8 waves.
__global__ void k_tiles(const float2* __restrict__ xy, const float* __restrict__ nrm,
                        float* __restrict__ tilePart) {
    __shared__ float2 sxy[128];
    __shared__ float  snrm[128];
    __shared__ float  sh[8];

    const int t    = threadIdx.x;
    const int lane = t & 31;
    const int wid  = t >> 5;
    const int row0 = blockIdx.y * 128 + wid * 16;
    const int colB = blockIdx.x * 128;

    // ---- async stage of the shared column panel (waves 0..3 only) ----
    if (t < 128) {
        uint32_t lds_xy = (uint32_t)(uintptr_t)(&sxy[t]);
        uint32_t lds_nr = (uint32_t)(uintptr_t)(&snrm[t]);
        const float2* g_xy = xy  + colB + t;
        const float*  g_nr = nrm + colB + t;
        asm volatile("global_load_async_to_lds_b64 %0, %1, off"
                     :: "v"(lds_xy), "v"(g_xy) : "memory");
        asm volatile("global_load_async_to_lds_b32 %0, %1, off"
                     :: "v"(lds_nr), "v"(g_nr) : "memory");
        asm volatile("s_wait_asynccnt 0x0" ::: "memory");
    }

    // ---- per-wave A fragment (16x4 f32, K=2,3 zero-padded) ----
    // Lanes 0-15 carry K=0,1 for row M=lane; lanes 16-31 carry the zero pad.
    const bool  lo = (lane < 16);
    const float2 av = xy[row0 + (lane & 15)];          // unconditional, no exec div.
    v2f a;
    a.x = lo ? av.x : 0.0f;
    a.y = lo ? av.y : 0.0f;

    // row norms for the C-layout rows this lane covers (M = r + 8*(lane>=16))
    const int hi = lo ? 0 : 8;
    v4f rn0 = *(const v4f*)(nrm + row0 + hi);          // rows hi..hi+3
    v4f rn1 = *(const v4f*)(nrm + row0 + hi + 4);      // rows hi+4..hi+7
    float rn[8] = { rn0.x, rn0.y, rn0.z, rn0.w, rn1.x, rn1.y, rn1.z, rn1.w };

    __syncthreads();   // LDS panel visible to all 8 waves

    float acc = 0.0f;
    #pragma unroll
    for (int ct = 0; ct < 8; ++ct) {
        const int ci = ct * 16 + (lane & 15);
        const float2 bv = sxy[ci];                     // ds_load_b64 (broadcast hi half)
        const float  cn = snrm[ci];                    // ds_load_b32
        v2f b;
        b.x = lo ? bv.x : 0.0f;
        b.y = lo ? bv.y : 0.0f;

        v8f c = {};
        c = __builtin_amdgcn_wmma_f32_16x16x4_f32(
                /*neg_a=*/false, a, /*neg_b=*/false, b,
                /*c_mod=*/(short)0, c, /*reuse_a=*/false, /*reuse_b=*/false);

        #pragma unroll
        for (int r = 0; r < 8; ++r) {
            float d2 = rn[r] + cn - 2.0f * c[r];       // = |xi-xj|^2
            acc += __builtin_amdgcn_rcpf(1.0f + d2);   // Student-t term
        }
    }

    acc = wave_reduce(acc);
    if (lane == 0) sh[wid] = acc;
    __syncthreads();
    if (t == 0) {
        float s = 0.f;
        #pragma unroll
        for (int k = 0; k < 8; ++k) s += sh[k];
        tilePart[blockIdx.y * 64 + blockIdx.x] = s;
    }
}

// --------------------------- K3: observed pairs -----------------------------
__global__ void k_pairs(const float* __restrict__ pij,
                        const int* __restrict__ ii, const int* __restrict__ jj,
                        const float2* __restrict__ xy,
                        float* __restrict__ s1Part, float* __restrict__ spPart) {
    __shared__ float sh[8];
    int t = blockIdx.x * 256 + threadIdx.x;
    int a = ii[t], b = jj[t];
    float2 pa = xy[a], pb = xy[b];
    float d0 = pb.x - pa.x + 1e-12f;
    float d1 = pb.y - pa.y + 1e-12f;
    float d2 = d0 * d0 + d1 * d1;
    float p  = pij[t];
    // log(num) = -log(1+d2); this contribution is independent of the partition
    float s1 = p * (logf(p + 1e-39f) + log1pf(d2));

    int lane = threadIdx.x & 31, wid = threadIdx.x >> 5;
    float r1 = wave_reduce(s1);
    float rp = wave_reduce(p);
    if (lane == 0) sh[wid] = r1;
    __syncthreads();
    if (threadIdx.x == 0) {
        float s = 0.f;
        #pragma unroll
        for (int k = 0; k < 8; ++k) s += sh[k];
        s1Part[blockIdx.x] = s;
    }
    __syncthreads();
    if (lane == 0) sh[wid] = rp;
    __syncthreads();
    if (threadIdx.x == 0) {
        float s = 0.f;
        #pragma unroll
        for (int k = 0; k < 8; ++k) s += sh[k];
        spPart[blockIdx.x] = s;
    }
}

// --------------------------- K4: final combine ------------------------------
__global__ void k_final(const float* __restrict__ kldPart, const float* __restrict__ tilePart,
                        const float* __restrict__ s1Part, const float* __restrict__ spPart,
                        float* __restrict__ out) {
    __shared__ float sh[8];
    __shared__ float res[3];
    int lane = threadIdx.x & 31, wid = threadIdx.x >> 5;

    // 1) partition sum over 4096 tile partials
    float tsum = 0.f;
    for (int k = threadIdx.x; k < TILE_BLKS; k += 256) tsum += tilePart[k];
    tsum = wave_reduce(tsum);
    if (lane == 0) sh[wid] = tsum;
    __syncthreads();
    if (threadIdx.x == 0) {
        float s = 0.f;
        #pragma unroll
        for (int k = 0; k < 8; ++k) s += sh[k];
        res[0] = s;
    }
    __syncthreads();

    // 2) pair sums (256 partials each)
    float s1 = s1Part[threadIdx.x];
    float sp = spPart[threadIdx.x];
    s1 = wave_reduce(s1);
    if (lane == 0) sh[wid] = s1;
    __syncthreads();
    if (threadIdx.x == 0) {
        float s = 0.f;
        #pragma unroll
        for (int k = 0; k < 8; ++k) s += sh[k];
        res[1] = s;
    }
    __syncthreads();
    sp = wave_reduce(sp);
    if (lane == 0) sh[wid] = sp;
    __syncthreads();
    if (threadIdx.x == 0) {
        float s = 0.f;
        #pragma unroll
        for (int k = 0; k < 8; ++k) s += sh[k];
        res[2] = s;
    }
    __syncthreads();

    // 3) KLD partials (32) + combine
    if (threadIdx.x == 0) {
        float kld = 0.f;
        #pragma unroll
        for (int k = 0; k < PTS_BLKS; ++k) kld += kldPart[k];
        float part       = res[0] - (float)N_POINTS;    // remove diagonal
        float loss_pairs = res[1] + res[2] * logf(part);
        float loss_kldrp = -0.5f * kld;
        // frac = 65536 / (67108864.0^2) = 2^-36 exactly
        out[0] = loss_pairs + loss_kldrp * 1.4551915228366852e-11f;
    }
}

// ---------------------------------------------------------------------------
extern "C" void kernel_launch(void* const* d_in, const int* in_sizes, int n_in,
                              void* d_out, int out_size, void* d_ws, size_t ws_size,
                              hipStream_t stream) {
    const float* pij = (const float*)d_in[0];
    const float* mu  = (const float*)d_in[1];
    const float* lv  = (const float*)d_in[2];
    const int*   ii  = (const int*)d_in[3];
    const int*   jj  = (const int*)d_in[4];
    float* out = (float*)d_out;

    float* W = (float*)d_ws;
    float2* xyv = (float2*)(W + WS_XY);
    float*  nrm = W + WS_NRM;
    float*  kldP = W + WS_KLD;
    float*  tilP = W + WS_TILE;
    float*  s1P  = W + WS_S1;
    float*  spP  = W + WS_SP;

    k_points<<<PTS_BLKS, 256, 0, stream>>>(mu, lv, xyv, nrm, kldP);
    k_tiles <<<dim3(64, 64), 256, 0, stream>>>(xyv, nrm, tilP);
    k_pairs <<<PAIR_BLKS, 256, 0, stream>>>(pij, ii, jj, xyv, s1P, spP);
    k_final <<<1, 256, 0, stream>>>(kldP, tilP, s1P, spP, out);
}